// GCNLayer_25829933318529
// MI455X (gfx1250) — compile-verified
//
#include <hip/hip_runtime.h>

typedef float v2f __attribute__((ext_vector_type(2)));
typedef float v8f __attribute__((ext_vector_type(8)));

#define IN_DIM 128
#define OUT_DIM 128

// ---------------------------------------------------------------------------
// Kernel 1: out[n][c] = bias[c]  (vectorized float4 broadcast init)
// ---------------------------------------------------------------------------
__global__ __launch_bounds__(256) void gcn_bias_init(
    float* __restrict__ out, const float* __restrict__ bias, long total4)
{
    long i = (long)blockIdx.x * blockDim.x + threadIdx.x;
    if (i < total4) {
        ((float4*)out)[i] = ((const float4*)bias)[i & 31];  // 128 floats = 32 float4
    }
}

// ---------------------------------------------------------------------------
// Kernel 2: H = X @ W via V_WMMA_F32_16X16X4_F32.
// Block = 256 threads = 8 waves; block covers 128 rows; each wave computes a
// 16-row strip across all 128 output columns (8 accumulators of 16x16 f32).
//
// W is staged once into LDS *pre-swizzled into the WMMA B-fragment register
// layout*: for k-block kb (K = 4*kb..4*kb+3), n-tile j, lane l, the two
// B elements that lane needs (K = kk, kk+1 at N = j*16 + l%16, kk = 2*(l/16))
// are stored contiguously. Each B operand is then ONE ds_load_b64 into an
// even VGPR pair: no v_mov shuffles, conflict-free banks (32 lanes x 8B
// consecutive = all 64 banks exactly once), and the whole 64 KB is reachable
// with 16-bit DS immediate offsets from a single base VGPR.
// ---------------------------------------------------------------------------
__global__ __launch_bounds__(256) void gcn_gemm_wmma(
    const float* __restrict__ X, const float* __restrict__ W,
    float* __restrict__ H, int n_nodes)
{
    __shared__ float Wlds[IN_DIM * OUT_DIM];   // 64 KB, swizzled layout

    const int tid = threadIdx.x;
    // Cooperative swizzled store: read W row-major (coalesced), scatter to
    // B-fragment layout: dst = ((kb*8 + j)*32 + lane)*2 + e
    for (int i = tid; i < IN_DIM * OUT_DIM; i += 256) {
        const int k  = i >> 7;         // 0..127
        const int nn = i & 127;        // 0..127
        const int kb   = k >> 2;       // k-block
        const int kr   = k & 3;        // K within block
        const int hlf  = kr >> 1;      // which lane half holds this K
        const int e    = kr & 1;       // element within the b64 pair
        const int j    = nn >> 4;      // n-tile
        const int ln   = (hlf << 4) | (nn & 15);
        Wlds[(((kb << 3) + j) << 6) + (ln << 1) + e] = W[i];
    }
    __syncthreads();

    const int wave = tid >> 5;
    const int lane = tid & 31;
    const int half = lane >> 4;        // 0: lanes 0-15, 1: lanes 16-31
    const int n    = lane & 15;
    const int kk   = half << 1;        // K sub-offset: 0 or 2
    const int row0 = blockIdx.x * 128 + wave * 16;

    // Clamp A row for loads: EXEC must stay all-1s for WMMA, so no predication.
    int arow = row0 + n;
    if (arow > n_nodes - 1) arow = n_nodes - 1;
    const float* __restrict__ xrow = X + (size_t)arow * IN_DIM;

    // Per-lane base into swizzled W
    const float* __restrict__ wbase = Wlds + (lane << 1);

    v8f acc[8];
    const v8f vzero = {0.f, 0.f, 0.f, 0.f, 0.f, 0.f, 0.f, 0.f};
#pragma unroll
    for (int j = 0; j < 8; ++j) acc[j] = vzero;

#pragma unroll 4
    for (int kb = 0; kb < IN_DIM / 4; ++kb) {
        // A fragment 16x4: VGPR0 = K=kk, VGPR1 = K=kk+1 (contiguous float2)
        const v2f a = *(const v2f*)(xrow + (kb << 2) + kk);
        const float* wk = wbase + (kb << 9);     // + kb*512 floats
#pragma unroll
        for (int j = 0; j < 8; ++j) {
            // B fragment: single contiguous 8B -> one ds_load_b64
            const v2f b = *(const v2f*)(wk + (j << 6));   // + j*64 floats
            acc[j] = __builtin_amdgcn_wmma_f32_16x16x4_f32(
                false, a, false, b, (short)0, acc[j], false, false);
        }
    }

    // C/D layout: VGPR r -> M = r (lanes 0-15) / r+8 (lanes 16-31), N = lane%16
    const int rbase = row0 + (half << 3);
#pragma unroll
    for (int j = 0; j < 8; ++j) {
#pragma unroll
        for (int r = 0; r < 8; ++r) {
            const int out_r = rbase + r;
            if (out_r < n_nodes)
                H[(size_t)out_r * OUT_DIM + j * 16 + n] = acc[j][r];
        }
    }
}

// ---------------------------------------------------------------------------
// Kernel 3: COO scatter  out[dst] += val * H[src].
// One wave per edge: 32 lanes x float4 = coalesced 512B gather of H[src],
// then 4 global_atomic_add_f32 per lane. Edge indices loaded 32-at-a-time
// (one coalesced load per 32 edges) and broadcast via v_readlane.
// h and out both fit in the 192 MB L2, so this is L2-throughput bound.
// ---------------------------------------------------------------------------
__global__ __launch_bounds__(256) void gcn_scatter(
    const float* __restrict__ H, const int* __restrict__ esrc,
    const int* __restrict__ edst, const float* __restrict__ eval,
    float* __restrict__ out, int n_edges)
{
    const int lane  = threadIdx.x & 31;
    const int wave  = (blockIdx.x * blockDim.x + threadIdx.x) >> 5;
    const int nwave = (gridDim.x * blockDim.x) >> 5;

    for (int e0 = wave * 32; e0 < n_edges; e0 += nwave * 32) {
        const int  eid = e0 + lane;
        const bool ok  = eid < n_edges;
        const int   sv = ok ? esrc[eid] : 0;
        const int   dv = ok ? edst[eid] : 0;
        const float wv = ok ? eval[eid] : 0.f;
        const int  cnt = (n_edges - e0 < 32) ? (n_edges - e0) : 32;

        for (int i = 0; i < cnt; ++i) {
            const int   s = __builtin_amdgcn_readlane(sv, i);
            const int   d = __builtin_amdgcn_readlane(dv, i);
            const float w = __builtin_bit_cast(
                float, __builtin_amdgcn_readlane(__builtin_bit_cast(int, wv), i));

            const float4 hv = ((const float4*)(H + (size_t)s * OUT_DIM))[lane];
            float* o = out + (size_t)d * OUT_DIM + lane * 4;
            atomicAdd(o + 0, hv.x * w);
            atomicAdd(o + 1, hv.y * w);
            atomicAdd(o + 2, hv.z * w);
            atomicAdd(o + 3, hv.w * w);
        }
    }
}

// ---------------------------------------------------------------------------
extern "C" void kernel_launch(void* const* d_in, const int* in_sizes, int n_in,
                              void* d_out, int out_size, void* d_ws, size_t ws_size,
                              hipStream_t stream)
{
    const float* X    = (const float*)d_in[0];
    const int*   esrc = (const int*)  d_in[1];
    const int*   edst = (const int*)  d_in[2];
    const float* ev   = (const float*)d_in[3];
    const float* W    = (const float*)d_in[4];
    const float* bias = (const float*)d_in[5];
    float* out = (float*)d_out;
    float* H   = (float*)d_ws;   // 100000*128*4 = 51.2 MB scratch

    const int n_nodes = in_sizes[0] / IN_DIM;
    const int n_edges = in_sizes[1];

    // 1) out = bias (broadcast)
    const long total4 = (long)n_nodes * (OUT_DIM / 4);
    gcn_bias_init<<<(int)((total4 + 255) / 256), 256, 0, stream>>>(out, bias, total4);

    // 2) H = X @ W   (WMMA f32 16x16x4, swizzled-LDS B fragments)
    gcn_gemm_wmma<<<(n_nodes + 127) / 128, 256, 0, stream>>>(X, W, H, n_nodes);

    // 3) out[dst] += val * H[src]
    const int groups = (n_edges + 31) / 32;     // one wave per 32-edge group
    const int blocks = (groups + 7) / 8;        // 8 waves per 256-thread block
    gcn_scatter<<<blocks, 256, 0, stream>>>(H, esrc, edst, ev, out, n_edges);
}